// Orbitals_ent_70832600645826
// MI455X (gfx1250) — compile-verified
//
#include <hip/hip_runtime.h>
#include <stdint.h>

// Reference: B=2048, N_SITES=256, N_ELECS=256, N_HID=64
// out[b, j, 0:256] = mf[src, :], out[b, j, 256:320] = hf[src, :]
// where src is the j-th smallest occupied spin-orbital index of batch b
// (top_k over the 512-wide occupancy mask == sorted occupied indices:
//  up sites ascending, then down sites ascending + 256).

typedef float    v4f __attribute__((ext_vector_type(4)));
typedef uint32_t su4 __attribute__((ext_vector_type(4)));   // SGPR quad
typedef uint32_t su8 __attribute__((ext_vector_type(8)));   // SGPR octet

#define NSITES      256
#define ROWS2       512                      // 2 * NSITES
#define NE          256                      // orbitals_mf cols
#define NH          64                       // orbitals_hf cols
#define ROWF        (NE + NH)                // 320 floats per output row
#define HF_LDS_B    (ROWS2 * NH * 4)         // 131072 bytes (full hf table)
#define SMEM_BYTES  (HF_LDS_B + NSITES * 4 + 64)

__global__ __launch_bounds__(256) void orb_gather_kernel(
    const int*   __restrict__ x,    // (B, 256) in {-1,+1}
    const float* __restrict__ mf,   // (512, 256)
    const float* __restrict__ hf,   // (512, 64)
    float*       __restrict__ out)  // (B, 256, 320)
{
    extern __shared__ char smem[];
    v4f* hf_lds  = (v4f*)smem;                          // 8192 x float4 = 128KB
    int* lds_src = (int*)(smem + HF_LDS_B);             // 256 ints: dest row -> src row
    int* lds_wt  = (int*)(smem + HF_LDS_B + NSITES*4);  // 8 per-wave up-counts

    const int b    = blockIdx.x;
    const int tid  = threadIdx.x;
    const int lane = tid & 31;
    const int wave = tid >> 5;

    // ---- TDM: stage the full hf table (512x64 f32 = 128KB) global->LDS with
    // one tensor_load_to_lds, issued by wave 0 only (wave-level DMA, EXEC
    // ignored). Tracked on TENSORcnt; overlaps with the scan + mf copy phases.
    if (tid < 32) {
        unsigned           lds_addr = (unsigned)(size_t)(void*)hf_lds;
        unsigned long long ga       = (unsigned long long)(const void*)hf;

        su4 g0;
        g0[0] = 1u;                                            // count=1 (valid D#)
        g0[1] = lds_addr;                                      // lds_addr[31:0]
        g0[2] = (uint32_t)ga;                                  // global_addr[31:0]
        g0[3] = (uint32_t)((ga >> 32) & 0x01FFFFFFull)         // global_addr[56:32]
              | (2u << 30);                                    // type=2 ("image")

        su8 g1;
        g1[0] = 2u << 16;                 // data_size=2 (4 bytes)
        g1[1] = (uint32_t)NH << 16;       // tensor_dim0 = 64 (bits 79:48, low half)
        g1[2] = (uint32_t)ROWS2 << 16;    // tensor_dim1 = 512 (bits 111:80, low half)
        g1[3] = (uint32_t)NH << 16;       // tile_dim0 = 64 (bits 127:112)
        g1[4] = (uint32_t)ROWS2;          // tile_dim1 = 512 (bits 143:128); tile_dim2=0
        g1[5] = (uint32_t)NH;             // tensor_dim0_stride = 64 (bits 207:160)
        g1[6] = 0u;                       // stride hi / tensor_dim1_stride lo
        g1[7] = 0u;                       // tensor_dim1_stride hi (unused, 2D)

        asm volatile("tensor_load_to_lds %0, %1" :: "s"(g0), "s"(g1) : "memory");
    }

    // ---- Occupancy scan (wave32 ballot + 8-wave prefix): sort-free top_k.
    __builtin_prefetch(mf, 0, 3);
    int  xv = x[b * NSITES + tid];
    bool up = (xv == 1);
    unsigned m  = __builtin_amdgcn_ballot_w32(up);
    int pre_in_wave = __builtin_popcount(m & ((1u << lane) - 1u));
    if (lane == 0) lds_wt[wave] = __builtin_popcount(m);
    __syncthreads();

    int wave_excl = 0, n_up = 0;
    #pragma unroll
    for (int w = 0; w < 8; ++w) {
        int t = lds_wt[w];
        wave_excl += (w < wave) ? t : 0;
        n_up      += t;
    }
    int pre = wave_excl + pre_in_wave;             // # up sites with index < tid
    int j   = up ? pre : (n_up + tid - pre);       // destination rank
    int src = up ? tid : (tid + NSITES);           // source spin-orbital row
    lds_src[j] = src;
    __syncthreads();

    // ---- mf copy: 256 rows x 64 float4. Loads are L2-resident (512KB table,
    // regular temporal hint keeps it cached); stores are non-temporal b128
    // (671MB write-once stream must not evict the tables from L2).
    float* outb = out + (size_t)b * (NSITES * ROWF);
    const v4f* __restrict__ mfv = (const v4f*)mf;  // row stride = 64 float4
    #pragma unroll 4
    for (int v = tid; v < NSITES * (NE / 4); v += 256) {
        int r  = v >> 6;                           // output row (wave-uniform)
        int c4 = v & 63;                           // float4 column
        int s  = lds_src[r];                       // broadcast ds_load
        v4f val = mfv[s * (NE / 4) + c4];
        __builtin_nontemporal_store(val, (v4f*)(outb + (size_t)r * ROWF + (c4 << 2)));
    }

    // ---- Drain the TDM transfer, then serve hf gathers from LDS (DS pipe;
    // keeps the VMEM pipe free for the NT store stream).
    __builtin_amdgcn_s_wait_tensorcnt(0);
    __syncthreads();

    #pragma unroll 4
    for (int v = tid; v < NSITES * (NH / 4); v += 256) {
        int r  = v >> 4;
        int c4 = v & 15;
        int s  = lds_src[r];
        v4f val = hf_lds[s * (NH / 4) + c4];
        __builtin_nontemporal_store(val, (v4f*)(outb + (size_t)r * ROWF + NE + (c4 << 2)));
    }
}

extern "C" void kernel_launch(void* const* d_in, const int* in_sizes, int n_in,
                              void* d_out, int out_size, void* d_ws, size_t ws_size,
                              hipStream_t stream) {
    const int*   x   = (const int*)d_in[0];
    const float* mf  = (const float*)d_in[1];
    const float* hf  = (const float*)d_in[2];
    float*       out = (float*)d_out;

    const int B = in_sizes[0] / NSITES;   // 2048

    (void)hipFuncSetAttribute((const void*)orb_gather_kernel,
                              hipFuncAttributeMaxDynamicSharedMemorySize, SMEM_BYTES);
    orb_gather_kernel<<<B, 256, SMEM_BYTES, stream>>>(x, mf, hf, out);
}